// MambaBlock_12378095747351
// MI455X (gfx1250) — compile-verified
//
#include <hip/hip_runtime.h>
#include <hip/hip_bf16.h>
#include <stdint.h>

// ---------------------------------------------------------------------------
// Mamba block for MI455X (gfx1250, wave32, WMMA, async double-buffered LDS).
// B=2, S=2048, HIDDEN=1024, MLP=2048, N_STATE=16, DT_RANK=64, CONV_K=4.
// ---------------------------------------------------------------------------

#define BB      2
#define SS      2048
#define HIDDEN  1024
#define MLP     2048
#define NSTATE  16
#define DTRANK  64
#define CONVK   4
#define MTOT    (BB * SS)              // 4096
#define XPW     (DTRANK + 2 * NSTATE)  // 96

typedef __attribute__((ext_vector_type(2))) float v2f;
typedef __attribute__((ext_vector_type(8))) float v8f;

#define BM   64       // block tile rows
#define BK   16       // K chunk
#define PADA 20       // sA row pitch (floats): 16B-aligned, conflict-free

// CDNA5 async memory->LDS copy (GLOBAL_LOAD_ASYNC_TO_LDS_B128, ASYNCcnt).
// lds_off = LDS byte address (low 32 bits of a __shared__ pointer),
// gptr    = per-lane global address.
__device__ __forceinline__ void async_copy_b128(uint32_t lds_off,
                                                const float* gptr) {
    asm volatile("global_load_async_to_lds_b128 %0, %1, off"
                 :: "v"(lds_off), "v"(gptr)
                 : "memory");
}
__device__ __forceinline__ void wait_async0() {
    asm volatile("s_wait_asynccnt 0x0" ::: "memory");
}

// ---------------------------------------------------------------------------
// LDS-tiled fp32 WMMA GEMM with async double buffering:
//   out[M x N] = act( A[M x K](row stride lda) @ W[K x N](row-major) + bias[N] )
// Block: 256 threads = 8 waves, tile BM x BN; wave (wm,wn) computes a
// 16 x (BN/2 / ... ) strip: NT = BN/32 16x16 tiles, reusing its A fragment.
//
// sA[2][BM][BK] pad PADA  (row-major)  -> A fragment: ds_load_b64
// sW[2][BK][BN] pitch BN+8 (row-major) -> B fragment: ds_load_2addr_b32
//   (pitch mod 32 == 8  =>  half-wave K rows 16 banks apart: conflict-free)
//
// fp32 WMMA fragment layout (ISA 7.12.2), lane L, ks = 2*(L/16):
//   a: m=L%16;  A[m][k0+ks+{0,1}]      b: n=L%16;  W[k0+ks+{0,1}][n]
//   D: acc[v] of lane L -> row v + 8*(L/16), col L%16
// act: 0 = identity, 1 = sigmoid, 2 = softplus
// ---------------------------------------------------------------------------
template <int BN>
__global__ __launch_bounds__(256)
void gemm_wmma_tiled(const float* __restrict__ A, int lda,
                     const float* __restrict__ W,
                     const float* __restrict__ bias,
                     float* __restrict__ out,
                     int N, int K, int act) {
    constexpr int NT   = BN / 32;       // 16-col tiles per wave (4 or 3)
    constexpr int HBN  = BN / 2;        // cols per wave group
    constexpr int NW4  = BK * BN / 4;   // float4s of W per chunk
    constexpr int PW   = BN + 8;        // sW row pitch (floats)
    constexpr int WIT  = (NW4 + 255) / 256;

    __shared__ float sA[2][BM * PADA];
    __shared__ float sW[2][BK * PW];

    const int tid  = threadIdx.x;
    const int lane = tid & 31;
    const int wid  = tid >> 5;
    const int wm   = wid & 3;           // 16-row strip
    const int wn   = wid >> 2;          // HBN-col group

    const int Nb = N / BN;
    const int bm = blockIdx.x / Nb;
    const int bn = blockIdx.x - bm * Nb;

    const int r    = lane & 15;
    const int half = lane >> 4;

    // staging coordinates (one async b128 per thread for A; WIT for W)
    const int arow  = tid >> 2;              // 0..63
    const int acol4 = (tid & 3) * 4;         // 0,4,8,12
    const float* __restrict__ aGlob =
        A + (size_t)(bm * BM + arow) * lda + acol4;

    auto stage = [&](int buf, int k0) {
        async_copy_b128((uint32_t)(uintptr_t)&sA[buf][arow * PADA + acol4],
                        aGlob + k0);
        #pragma unroll
        for (int i = 0; i < WIT; ++i) {
            const int idx = tid + i * 256;
            if (NW4 % 256 == 0 || idx < NW4) {       // wave-uniform split
                const int wk = idx / (BN / 4);       // 0..15
                const int n4 = (idx % (BN / 4)) * 4; // col group
                async_copy_b128((uint32_t)(uintptr_t)&sW[buf][wk * PW + n4],
                                W + (size_t)(k0 + wk) * N + bn * BN + n4);
            }
        }
    };

    v8f acc[NT];
    #pragma unroll
    for (int i = 0; i < NT; ++i) acc[i] = (v8f){};

    const int nch = K / BK;
    stage(0, 0);                                   // prologue: chunk 0
    for (int ci = 0; ci < nch; ++ci) {
        const int cur = ci & 1;
        wait_async0();          // my copies for chunk ci have landed
        __syncthreads();        // everyone's copies landed; prev compute done
        if (ci + 1 < nch) stage(cur ^ 1, (ci + 1) * BK);

        #pragma unroll
        for (int kk = 0; kk < BK; kk += 4) {
            const int ko = kk + 2 * half;
            const v2f a = *(const v2f*)&sA[cur][(wm * 16 + r) * PADA + ko];
            #pragma unroll
            for (int nt = 0; nt < NT; ++nt) {
                const float* wb = &sW[cur][ko * PW + wn * HBN + nt * 16 + r];
                v2f b;  b.x = wb[0];  b.y = wb[PW];
                acc[nt] = __builtin_amdgcn_wmma_f32_16x16x4_f32(
                    false, a, false, b, (short)0, acc[nt], false, false);
            }
        }
        __syncthreads();        // compute(ci) done before buffer reuse
    }

    // ---- epilogue --------------------------------------------------------
    const int rowBase = bm * BM + wm * 16 + 8 * half;
    #pragma unroll
    for (int nt = 0; nt < NT; ++nt) {
        const int col = bn * BN + wn * HBN + nt * 16 + r;
        const float bv = bias ? bias[col] : 0.0f;
        #pragma unroll
        for (int v = 0; v < 8; ++v) {
            float x = acc[nt][v] + bv;
            if (act == 1)      x = 1.0f / (1.0f + __expf(-x));
            else if (act == 2) x = (x > 20.0f) ? x : log1pf(__expf(x));
            out[(size_t)(rowBase + v) * N + col] = x;
        }
    }
}

// ---------------------------------------------------------------------------
// Causal depthwise conv (K=4) + sigmoid.  One thread per (b,t,d).
// ---------------------------------------------------------------------------
__global__ __launch_bounds__(256)
void conv_sigmoid(const float* __restrict__ u,
                  const float* __restrict__ cw,   // (MLP, CONVK)
                  const float* __restrict__ cb,
                  float* __restrict__ y) {
    const int idx = blockIdx.x * 256 + threadIdx.x;
    const int d   = idx & (MLP - 1);
    const int bt  = idx >> 11;                    // MLP == 2^11
    const int t   = bt & (SS - 1);

    float acc = cb[d];
    #pragma unroll
    for (int k = 0; k < CONVK; ++k) {
        const int ts = t - (CONVK - 1) + k;
        if (ts >= 0)
            acc += u[(size_t)(bt - (CONVK - 1) + k) * MLP + d] * cw[d * CONVK + k];
    }
    y[(size_t)bt * MLP + d] = 1.0f / (1.0f + __expf(-acc));
}

// ---------------------------------------------------------------------------
// Selective scan: one thread per (batch, channel); 16 states in registers;
// B_t/C_t staged through LDS in chunks of TCHUNK timesteps.
// Output y*gate overwrites `gate` in place.
// ---------------------------------------------------------------------------
#define TCHUNK 8

__global__ __launch_bounds__(256)
void selective_scan(const float* __restrict__ u,
                    const float* __restrict__ delta,
                    const float* __restrict__ xp,      // (B,S,96)
                    const float* __restrict__ A_log,   // (MLP,16)
                    const float* __restrict__ D_skip,
                    float* __restrict__ gate_io) {
    const int tid = threadIdx.x;
    const int blkPerB = MLP / 256;
    const int b = blockIdx.x / blkPerB;
    const int d = (blockIdx.x % blkPerB) * 256 + tid;

    float A[NSTATE], h[NSTATE];
    #pragma unroll
    for (int n = 0; n < NSTATE; ++n) {
        A[n] = -__expf(A_log[d * NSTATE + n]);
        h[n] = 0.0f;
    }
    const float dsk = D_skip[d];

    __shared__ float sBC[TCHUNK][2 * NSTATE];

    for (int t0 = 0; t0 < SS; t0 += TCHUNK) {
        __syncthreads();
        {
            const int tt = tid >> 5, j = tid & 31;
            sBC[tt][j] = xp[(size_t)(b * SS + t0 + tt) * XPW + DTRANK + j];
        }
        __syncthreads();

        for (int tt = 0; tt < TCHUNK; ++tt) {
            const size_t base = (size_t)(b * SS + t0 + tt) * MLP + d;
            const float dlt = delta[base];
            const float uv  = u[base];
            float y = 0.0f;
            #pragma unroll
            for (int n = 0; n < NSTATE; ++n) {
                h[n] = __expf(dlt * A[n]) * h[n] + dlt * sBC[tt][n] * uv;
                y += h[n] * sBC[tt][NSTATE + n];
            }
            gate_io[base] = (y + uv * dsk) * gate_io[base];
        }
    }
}

// ---------------------------------------------------------------------------
// Launch.  Inputs: 0:x 1:W1 2:b1 3:W2 4:b2 5:conv_w 6:conv_b 7:A_log
// 8:x_proj_w 9:dt_proj_w 10:dt_proj_b 11:D_skip 12:Wo 13:bo
// Workspace (floats): gate | u_pre->delta (aliased) | u_conv | xp
// ---------------------------------------------------------------------------
extern "C" void kernel_launch(void* const* d_in, const int* in_sizes, int n_in,
                              void* d_out, int out_size, void* d_ws, size_t ws_size,
                              hipStream_t stream) {
    const float* x    = (const float*)d_in[0];
    const float* W1   = (const float*)d_in[1];
    const float* b1   = (const float*)d_in[2];
    const float* W2   = (const float*)d_in[3];
    const float* b2   = (const float*)d_in[4];
    const float* cw   = (const float*)d_in[5];
    const float* cb   = (const float*)d_in[6];
    const float* Alog = (const float*)d_in[7];
    const float* xpw  = (const float*)d_in[8];
    const float* dtw  = (const float*)d_in[9];
    const float* dtb  = (const float*)d_in[10];
    const float* Dsk  = (const float*)d_in[11];
    const float* Wo   = (const float*)d_in[12];
    const float* bo   = (const float*)d_in[13];
    float* out = (float*)d_out;

    float* gate  = (float*)d_ws;
    float* upre  = gate  + (size_t)MTOT * MLP;
    float* uconv = upre  + (size_t)MTOT * MLP;
    float* xp    = uconv + (size_t)MTOT * MLP;
    float* delta = upre;                     // alias: u_pre dead after conv

    const dim3 blk(256);
    const int gBig = (MTOT / BM) * (MLP / 128);     // 64*16 = 1024
    const int gXp  = (MTOT / BM) * (XPW / 96);      // 64
    const int gOut = (MTOT / BM) * (HIDDEN / 128);  // 512

    // gate = sigmoid(x @ W2 + b2)
    gemm_wmma_tiled<128><<<gBig, blk, 0, stream>>>(x, HIDDEN, W2, b2, gate,
                                                   MLP, HIDDEN, 1);
    // u_pre = x @ W1 + b1
    gemm_wmma_tiled<128><<<gBig, blk, 0, stream>>>(x, HIDDEN, W1, b1, upre,
                                                   MLP, HIDDEN, 0);
    // u = sigmoid(causal_conv(u_pre))
    conv_sigmoid<<<(MTOT * MLP) / 256, blk, 0, stream>>>(upre, cw, cb, uconv);
    // xp = u @ x_proj_w     (N = 96)
    gemm_wmma_tiled<96><<<gXp, blk, 0, stream>>>(uconv, MLP, xpw, nullptr, xp,
                                                 XPW, MLP, 0);
    // delta = softplus(xp[:, :64] @ dt_proj_w + dt_proj_b)   (lda = 96)
    gemm_wmma_tiled<128><<<gBig, blk, 0, stream>>>(xp, XPW, dtw, dtb, delta,
                                                   MLP, DTRANK, 2);
    // scan: gate <- (scan(u, delta, B, C) + u*D) * gate
    selective_scan<<<BB * (MLP / 256), blk, 0, stream>>>(uconv, delta, xp,
                                                         Alog, Dsk, gate);
    // out = y_gated @ Wo + bo
    gemm_wmma_tiled<128><<<gOut, blk, 0, stream>>>(gate, MLP, Wo, bo, out,
                                                   HIDDEN, MLP, 0);
}